// Model_20143396618767
// MI455X (gfx1250) — compile-verified
//
#include <hip/hip_runtime.h>
#include <hip/hip_bf16.h>
#include <math.h>

// ---- problem constants (match reference) ----
constexpr int NN    = 862;          // nodes
constexpr int NP    = 864;          // padded nodes (multiple of 32 for WMMA tiling)
constexpr int LSEQ  = 336;          // seq len
constexpr int LP    = 352;          // padded seq len (11 tiles of 32)
constexpr int PRED  = 336;          // pred len
constexpr int PER   = 24;           // period
constexpr int SEGX  = 14;           // 336/24
constexpr int SEGY  = 14;
constexpr int KS    = 25;           // conv kernel size
constexpr int PADK  = 12;
constexpr int EMB   = 40;
constexpr int KTOP  = 20;
constexpr int BATCH = 64;
constexpr float ALPHA_C = 3.0f;
constexpr int TN    = 32;           // node tile for LDS transpose kernel

typedef __attribute__((ext_vector_type(2))) float v2f;
typedef __attribute__((ext_vector_type(8))) float v8f;

// -------------------------------------------------------------------------
// Kernel 1: center + K=25 conv residual. Input x is [B, L, N]; output z kept
// in TRANSPOSED layout [B, LP, NP] (l-major) so WMMA B-fragments and the
// epilogue access contiguous memory. Pads (n in [862,864), l in [336,352))
// are zero-filled. Also emits per-(b,n) mean.
// -------------------------------------------------------------------------
__global__ __launch_bounds__(256) void center_conv_kernel(
    const float* __restrict__ x, const float* __restrict__ conv_w,
    float* __restrict__ z, float* __restrict__ meanOut) {
  __shared__ float tile[TN][LSEQ + 1];   // +1: odd stride -> bank-conflict free
  __shared__ float partial[TN][8];
  __shared__ float rowMean[TN];
  __shared__ float w[KS];

  const int tid = threadIdx.x;
  const int b   = blockIdx.y;
  const int n0  = blockIdx.x * TN;

  if (tid < KS) w[tid] = conv_w[tid];

  // coalesced transpose load: consecutive tid -> consecutive n
  for (int idx = tid; idx < TN * LSEQ; idx += 256) {
    int l = idx / TN, j = idx % TN;
    int n = n0 + j;
    tile[j][l] = (n < NN) ? x[((size_t)b * LSEQ + l) * NN + n] : 0.0f;
  }
  __syncthreads();

  // row means (8 partial lanes per row)
  {
    int j = tid >> 3, s = tid & 7;
    float acc = 0.0f;
    for (int l = s; l < LSEQ; l += 8) acc += tile[j][l];
    partial[j][s] = acc;
  }
  __syncthreads();
  if (tid < TN) {
    float acc = 0.0f;
    #pragma unroll
    for (int s = 0; s < 8; ++s) acc += partial[tid][s];
    float mval = acc * (1.0f / (float)LSEQ);
    rowMean[tid] = mval;
    int n = n0 + tid;
    if (n < NN) meanOut[(size_t)b * NN + n] = mval;
  }
  __syncthreads();

  // center in place
  for (int idx = tid; idx < TN * LSEQ; idx += 256) {
    int j = idx / LSEQ, l = idx % LSEQ;
    tile[j][l] -= rowMean[j];
  }
  __syncthreads();

  // conv (cross-correlation, zero-padded) + residual; write z[b, l, n]
  for (int idx = tid; idx < TN * LSEQ; idx += 256) {
    int l = idx / TN, j = idx % TN;      // consecutive tid -> consecutive n
    float s = tile[j][l];
    #pragma unroll
    for (int k = 0; k < KS; ++k) {
      int ls = l + k - PADK;
      if (ls >= 0 && ls < LSEQ) s += w[k] * tile[j][ls];
    }
    z[((size_t)b * LP + l) * NP + (n0 + j)] = s;
  }
  // zero the l-pad rows [336, 352)
  for (int idx = tid; idx < TN * (LP - LSEQ); idx += 256) {
    int l = LSEQ + idx / TN, j = idx % TN;
    z[((size_t)b * LP + l) * NP + (n0 + j)] = 0.0f;
  }
}

// -------------------------------------------------------------------------
// Kernel 2: nv = tanh(ALPHA * (emb @ lin_w^T + lin_b))   [N, EMB]
// -------------------------------------------------------------------------
__global__ __launch_bounds__(64) void nv_kernel(
    const float* __restrict__ emb, const float* __restrict__ lw,
    const float* __restrict__ lb, float* __restrict__ nv) {
  int n = blockIdx.x;
  int d = threadIdx.x;
  if (d >= EMB) return;
  float s = lb[d];
  #pragma unroll 8
  for (int e = 0; e < EMB; ++e) s += emb[n * EMB + e] * lw[d * EMB + e];
  nv[n * EMB + d] = tanhf(ALPHA_C * s);
}

// -------------------------------------------------------------------------
// Kernel 3: adjacency row build + top-20 sparsify (iterative max extraction,
// index tie-break = lowest index, matching lax.top_k). Padded to [NP,NP].
// -------------------------------------------------------------------------
__global__ __launch_bounds__(256) void adj_build_kernel(
    const float* __restrict__ nv1, const float* __restrict__ nv2,
    float* __restrict__ adj) {
  __shared__ float row[NN];
  __shared__ float u1[EMB], u2[EMB];
  __shared__ float redV[256];
  __shared__ int   redI[256];

  const int u = blockIdx.x;
  const int tid = threadIdx.x;

  for (int v = tid; v < NP; v += 256) adj[(size_t)u * NP + v] = 0.0f;
  if (u >= NN) return;  // pad rows stay zero (uniform branch per block)

  if (tid < EMB) { u1[tid] = nv1[u * EMB + tid]; u2[tid] = nv2[u * EMB + tid]; }
  __syncthreads();

  for (int v = tid; v < NN; v += 256) {
    float s = 0.0f;
    #pragma unroll 8
    for (int e = 0; e < EMB; ++e)
      s += u1[e] * nv2[v * EMB + e] - u2[e] * nv1[v * EMB + e];
    float t = tanhf(ALPHA_C * s);
    row[v] = (t > 0.0f) ? t : 0.0f;
  }
  __syncthreads();

  for (int iter = 0; iter < KTOP; ++iter) {
    float bv = -1.0f; int bi = NN;
    for (int v = tid; v < NN; v += 256) {
      float val = row[v];
      if (val > bv) { bv = val; bi = v; }   // ascending scan keeps lowest index
    }
    redV[tid] = bv; redI[tid] = bi;
    __syncthreads();
    for (int off = 128; off > 0; off >>= 1) {
      if (tid < off) {
        float ov = redV[tid + off]; int oi = redI[tid + off];
        if (ov > redV[tid] || (ov == redV[tid] && oi < redI[tid])) {
          redV[tid] = ov; redI[tid] = oi;
        }
      }
      __syncthreads();
    }
    if (tid == 0) {
      int sel = redI[0];
      adj[(size_t)u * NP + sel] = redV[0];
      row[sel] = -2.0f;  // consume
    }
    __syncthreads();
  }
}

// -------------------------------------------------------------------------
// Kernel 4: one Euler step  zout = (1-h)*zin + h*(adj @ zin)  via fp32 WMMA,
// with z stored [B, LP, NP] (l-major).
// One wave per block; 2x2 register blocking: 2 u-tiles x 2 l-tiles, so each
// A/B fragment (one b64 load each) feeds two of the four WMMAs, and there
// are 4 independent accumulation chains.
// A-frag (16x4 f32): lane m=lane&15 -> row; lane-half -> K {0,1}|{2,3}.
// B-frag (4x16 f32): lane m -> col (l);  lane-half -> K rows {0,1}|{2,3};
//   in [L,NP] layout the two K elements per lane are adjacent -> b64.
// C/D (16x16 f32, 8 VGPRs): vgpr r -> row r + 8*(lane>=16), col = lane&15;
//   in [L,NP] layout each lane's 8 results are 8 contiguous floats.
// -------------------------------------------------------------------------
__global__ __launch_bounds__(32) void ode_step_kernel(
    const float* __restrict__ adj, const float* __restrict__ zin,
    float* __restrict__ zout) {
  const int lane = threadIdx.x;
  const int m  = lane & 15;
  const int hi = lane >> 4;
  const int u0 = blockIdx.x * 32;
  const int u1 = u0 + 16;
  const int l0 = blockIdx.y * 32;
  const int l1 = l0 + 16;
  const int b  = blockIdx.z;

  const float* zb = zin  + (size_t)b * LP * NP;
  float*       zo = zout + (size_t)b * LP * NP;

  v8f acc00 = {}, acc01 = {}, acc10 = {}, acc11 = {};

  const size_t rowA0 = (size_t)(u0 + m) * NP;
  const size_t rowA1 = (size_t)(u1 + m) * NP;
  const size_t rowB0 = (size_t)(l0 + m) * NP;
  const size_t rowB1 = (size_t)(l1 + m) * NP;

  for (int v0 = 0; v0 < NP; v0 += 4) {
    const int ka = v0 + 2 * hi;                 // even -> 8B aligned
    v2f a0 = *(const v2f*)(adj + rowA0 + ka);
    v2f a1 = *(const v2f*)(adj + rowA1 + ka);
    v2f b0 = *(const v2f*)(zb  + rowB0 + ka);
    v2f b1 = *(const v2f*)(zb  + rowB1 + ka);
    acc00 = __builtin_amdgcn_wmma_f32_16x16x4_f32(
        false, a0, false, b0, (short)0, acc00, false, false);
    acc01 = __builtin_amdgcn_wmma_f32_16x16x4_f32(
        false, a0, false, b1, (short)0, acc01, false, false);
    acc10 = __builtin_amdgcn_wmma_f32_16x16x4_f32(
        false, a1, false, b0, (short)0, acc10, false, false);
    acc11 = __builtin_amdgcn_wmma_f32_16x16x4_f32(
        false, a1, false, b1, (short)0, acc11, false, false);
  }

  const float h  = 0.1f;          // ODE_T / ODE_STEPS
  const float om = 1.0f - h;

  // each (tile) epilogue: lane owns 8 contiguous floats starting at
  // (l + m)*NP + u + 8*hi  -> merges to wide load/store
  {
    const size_t base = (size_t)(l0 + m) * NP + u0 + 8 * hi;
    #pragma unroll
    for (int r = 0; r < 8; ++r) zo[base + r] = om * zb[base + r] + h * acc00[r];
  }
  {
    const size_t base = (size_t)(l1 + m) * NP + u0 + 8 * hi;
    #pragma unroll
    for (int r = 0; r < 8; ++r) zo[base + r] = om * zb[base + r] + h * acc01[r];
  }
  {
    const size_t base = (size_t)(l0 + m) * NP + u1 + 8 * hi;
    #pragma unroll
    for (int r = 0; r < 8; ++r) zo[base + r] = om * zb[base + r] + h * acc10[r];
  }
  {
    const size_t base = (size_t)(l1 + m) * NP + u1 + 8 * hi;
    #pragma unroll
    for (int r = 0; r < 8; ++r) zo[base + r] = om * zb[base + r] + h * acc11[r];
  }
}

// -------------------------------------------------------------------------
// Kernel 5: segment remap (14 -> 14 per period phase) + mean add.
// With z in [B, LP, NP] layout, reads AND writes are natively coalesced
// over n -- no LDS transpose needed.
// -------------------------------------------------------------------------
__global__ __launch_bounds__(256) void seg_out_kernel(
    const float* __restrict__ z, const float* __restrict__ meanIn,
    const float* __restrict__ lin_w, float* __restrict__ out) {
  __shared__ float w[SEGY * SEGX];
  const int tid = threadIdx.x;
  if (tid < SEGY * SEGX) w[tid] = lin_w[tid];
  __syncthreads();

  const int n = blockIdx.x * 256 + tid;
  const int t = blockIdx.y;
  const int b = blockIdx.z;
  if (n >= NN) return;

  const int p  = t % PER;
  const int sy = t / PER;
  float s = meanIn[(size_t)b * NN + n];
  #pragma unroll
  for (int sx = 0; sx < SEGX; ++sx)
    s += z[((size_t)b * LP + (sx * PER + p)) * NP + n] * w[sy * SEGX + sx];
  out[((size_t)b * PRED + t) * NN + n] = s;
}

// -------------------------------------------------------------------------
extern "C" void kernel_launch(void* const* d_in, const int* in_sizes, int n_in,
                              void* d_out, int out_size, void* d_ws, size_t ws_size,
                              hipStream_t stream) {
  (void)in_sizes; (void)n_in; (void)out_size; (void)ws_size;
  const float* x      = (const float*)d_in[0];
  const float* conv_w = (const float*)d_in[1];
  const float* emb1   = (const float*)d_in[2];
  const float* emb2   = (const float*)d_in[3];
  const float* lin1_w = (const float*)d_in[4];
  const float* lin1_b = (const float*)d_in[5];
  const float* lin2_w = (const float*)d_in[6];
  const float* lin2_b = (const float*)d_in[7];
  const float* lin_w  = (const float*)d_in[8];
  float* out = (float*)d_out;

  // workspace layout (floats)
  float* ws = (float*)d_ws;
  const size_t SZ = (size_t)BATCH * LP * NP;          // 19,464,192 floats
  float* zA      = ws;
  float* zB      = zA + SZ;
  float* adj     = zB + SZ;                           // NP*NP
  float* meanBuf = adj + (size_t)NP * NP;             // BATCH*NN
  float* nv1     = meanBuf + (size_t)BATCH * NN;      // NN*EMB
  float* nv2     = nv1 + (size_t)NN * EMB;            // NN*EMB

  dim3 gridTile(NP / TN, BATCH);                      // 27 x 64
  center_conv_kernel<<<gridTile, 256, 0, stream>>>(x, conv_w, zA, meanBuf);

  nv_kernel<<<NN, 64, 0, stream>>>(emb1, lin1_w, lin1_b, nv1);
  nv_kernel<<<NN, 64, 0, stream>>>(emb2, lin2_w, lin2_b, nv2);
  adj_build_kernel<<<NP, 256, 0, stream>>>(nv1, nv2, adj);

  dim3 gridODE(NP / 32, LP / 32, BATCH);              // 27 x 11 x 64 waves
  for (int s = 0; s < 10; ++s) {
    const float* zi = (s & 1) ? zB : zA;
    float*       zo = (s & 1) ? zA : zB;
    ode_step_kernel<<<gridODE, 32, 0, stream>>>(adj, zi, zo);
  }
  // after an even number of steps, result lives in zA

  dim3 gridOut((NN + 255) / 256, PRED, BATCH);
  seg_out_kernel<<<gridOut, 256, 0, stream>>>(zA, meanBuf, lin_w, out);
}